// StrokeRenderer_10290741641478
// MI455X (gfx1250) — compile-verified
//
#include <hip/hip_runtime.h>

typedef __attribute__((ext_vector_type(2))) float v2f;
typedef __attribute__((ext_vector_type(8))) float v8f;

#define GSZ 256
#define NCURVE 16
#define NPTS 16
#define NSAMP 10
#define PTS_PER_STROKE (150 + 1)          // (K-1)*P + final point
#define NSEG_PER_STROKE 150
#define NSEG_TOTAL (NCURVE * NSEG_PER_STROKE)   // 2400
#define TWID 5.0f
#define EPS 1e-5f

// ---------------------------------------------------------------------------
// Kernel 1: Bezier sampling via V_WMMA_F32_16X16X4_F32.
// One wave (32 lanes) per (stroke s, dim d): D[16x16] = A[16x4] * B[4x16]
//   A row r (segment) = [pt_r, pt_r+dv_r, pt_{r+1}-dv_{r+1}, pt_{r+1}] (dim d)
//   B col p (sample)  = bezier weights w1..w4 at t=p (p<10; cols 10..15 = 0)
// A layout (ISA 32-bit A 16x4): lane<16 holds {K0,K1}, lane>=16 holds {K2,K3}.
// B layout mirrored: lane<16 holds rows {K0,K1} at N=lane; lane>=16 rows {K2,K3}.
// D layout: lanes 0-15 -> M=j (vgpr j), lanes 16-31 -> M=j+8.
// ---------------------------------------------------------------------------
__global__ void bezier_sample_kernel(const float* __restrict__ curves,
                                     float* __restrict__ strokes) {
  const int s    = blockIdx.x >> 1;   // stroke
  const int d    = blockIdx.x & 1;    // coordinate dim (x/y)
  const int lane = threadIdx.x;       // 0..31
  const int r    = lane & 15;         // segment row for A / sample col for B

  // ---- Build A fragment -------------------------------------------------
  float a0 = 0.0f, a1 = 0.0f;
  if (r < NSEG_PER_STROKE - 135) { /* false for r == 15; empty body, no effect */ }
  if (r < 15) {
    if (lane < 16) {
      const float* cv = curves + ((s * NPTS + r) * 4);
      const float p  = cv[d];
      const float dv = cv[2 + d];
      a0 = p;        // K=0 : pts[r]
      a1 = p + dv;   // K=1 : after[r]
    } else {
      const float* cv = curves + ((s * NPTS + r + 1) * 4);
      const float p  = cv[d];
      const float dv = cv[2 + d];
      a0 = p - dv;   // K=2 : before[r+1]
      a1 = p;        // K=3 : pts[r+1]
    }
  }

  // ---- Build B fragment (Bezier basis, P = 10) --------------------------
  float b0 = 0.0f, b1 = 0.0f;
  if (r < NSAMP) {
    const float t   = (float)r;
    const float pf  = (float)NSAMP;
    const float u   = pf - t;
    const float inv = 1.0f / (pf * pf * pf);
    if (lane < 16) {
      b0 = u * u * u * inv;          // K=0 : w1
      b1 = 3.0f * u * u * t * inv;   // K=1 : w2
    } else {
      b0 = 3.0f * u * t * t * inv;   // K=2 : w3
      b1 = t * t * t * inv;          // K=3 : w4
    }
  }

  v2f A; A[0] = a0; A[1] = a1;
  v2f B; B[0] = b0; B[1] = b1;
  v8f C = {};
  // D = A x B + 0   (emits v_wmma_f32_16x16x4_f32)
  v8f D = __builtin_amdgcn_wmma_f32_16x16x4_f32(
      /*neg_a=*/false, A, /*neg_b=*/false, B,
      /*c_mod=*/(short)0, C, /*reuse_a=*/false, /*reuse_b=*/false);

  // ---- Scatter D to strokes[s][idx][d] * G ------------------------------
  const int pcol = lane & 15;
  if (pcol < NSAMP) {
#pragma unroll
    for (int j = 0; j < 8; ++j) {
      const int row = (lane < 16) ? j : (j + 8);  // segment index
      if (row < 15) {
        const int idx = row * NSAMP + pcol;       // 0..149
        strokes[(s * PTS_PER_STROKE + idx) * 2 + d] = D[j] * (float)GSZ;
      }
    }
  }
  // Final point of the stroke: pts[s][15][d] * G
  if (lane == 0) {
    const float* cv = curves + ((s * NPTS + (NPTS - 1)) * 4);
    strokes[(s * PTS_PER_STROKE + (PTS_PER_STROKE - 1)) * 2 + d] =
        cv[d] * (float)GSZ;
  }
}

// ---------------------------------------------------------------------------
// Kernel 2: distance-field render. One thread per pixel, segment table in LDS.
// Inner loop tracks min squared distance; sqrt + /T hoisted out of the loop.
// ---------------------------------------------------------------------------
__global__ __launch_bounds__(256) void render_kernel(
    const float* __restrict__ strokes, float* __restrict__ out) {
  __shared__ float4 segv[NSEG_TOTAL];   // vx, vy, dx, dy
  __shared__ float  seginv[NSEG_TOTAL]; // 1 / (dx^2 + dy^2 + eps)

  // Cooperative build of the per-segment table (same for every block; the
  // 19 KB stroke buffer is L2-resident so this is cheap).
  for (int i = threadIdx.x; i < NSEG_TOTAL; i += blockDim.x) {
    const int s = i / NSEG_PER_STROKE;
    const int k = i - s * NSEG_PER_STROKE;
    const float* v = strokes + (s * PTS_PER_STROKE + k) * 2;
    const float vx = v[0], vy = v[1];
    const float wx = v[2], wy = v[3];   // next point is contiguous
    const float dx = wx - vx, dy = wy - vy;
    const float d2 = dx * dx + dy * dy;
    segv[i]   = make_float4(vx, vy, dx, dy);
    seginv[i] = 1.0f / (d2 + EPS);
  }
  __syncthreads();

  const int id = blockIdx.x * blockDim.x + threadIdx.x; // x*G + y
  const float px = (float)(id >> 8);   // xc (indexing='ij')
  const float py = (float)(id & 255);  // yc

  float mind2 = 1e30f;
#pragma unroll 4
  for (int i = 0; i < NSEG_TOTAL; ++i) {
    const float4 sv = segv[i];
    const float rx = px - sv.x;
    const float ry = py - sv.y;
    float t = (rx * sv.z + ry * sv.w) * seginv[i];
    t = fminf(fmaxf(t, 0.0f), 1.0f);
    const float ex = rx - t * sv.z;
    const float ey = ry - t * sv.w;
    const float d2p = ex * ex + ey * ey;
    mind2 = fminf(mind2, d2p);
  }

  const float darkness = sqrtf(mind2) * (1.0f / TWID);
  out[id] = fminf(1.0f, darkness);
}

extern "C" void kernel_launch(void* const* d_in, const int* in_sizes, int n_in,
                              void* d_out, int out_size, void* d_ws, size_t ws_size,
                              hipStream_t stream) {
  const float* curves = (const float*)d_in[0];   // (16, 16, 4) float32
  float* strokes = (float*)d_ws;                 // 16 * 151 * 2 floats
  float* out = (float*)d_out;                    // 256 * 256 floats

  // 32 waves: one (stroke, dim) pair each -> one v_wmma_f32_16x16x4_f32 each.
  bezier_sample_kernel<<<NCURVE * 2, 32, 0, stream>>>(curves, strokes);
  // 256 pixels per block, 256 blocks.
  render_kernel<<<(GSZ * GSZ) / 256, 256, 0, stream>>>(strokes, out);
}